// FilterBankConstructorND_27401891349296
// MI455X (gfx1250) — compile-verified
//
#include <hip/hip_runtime.h>
#include <cstdint>

#define TPB 256

typedef __attribute__((ext_vector_type(2))) float v2f;
typedef __attribute__((ext_vector_type(8))) float v8f;
typedef __attribute__((ext_vector_type(4))) int v4i;

#define AS1 __attribute__((address_space(1)))
#define AS3 __attribute__((address_space(3)))

#ifndef __has_builtin
#define __has_builtin(x) 0
#endif

#if __has_builtin(__builtin_amdgcn_global_load_async_to_lds_b128) && \
    __has_builtin(__builtin_amdgcn_global_load_async_to_lds_b32)
#define USE_ASYNC_LOAD 1
#else
#define USE_ASYNC_LOAD 0
#endif

#if __has_builtin(__builtin_amdgcn_global_store_async_from_lds_b128) && \
    __has_builtin(__builtin_amdgcn_global_store_async_from_lds_b32)
#define USE_ASYNC_STORE 1
#else
#define USE_ASYNC_STORE 0
#endif

#if __has_builtin(__builtin_amdgcn_wmma_f32_16x16x4_f32)
#define USE_WMMA 1
#else
#define USE_WMMA 0
#endif

__device__ __forceinline__ unsigned lds_off(const void* p) {
  return (unsigned)(unsigned long long)(const AS3 char*)p;
}

__device__ __forceinline__ void wait_async0() {
#if __has_builtin(__builtin_amdgcn_s_wait_asynccnt)
  __builtin_amdgcn_s_wait_asynccnt(0);
#else
  asm volatile("s_wait_asynccnt 0x0" ::: "memory");
#endif
}

// Cross-half-wave swap: lane i <-> lane i^16. Prefer VALU v_permlanex16_b32
// (identity lane-select) over any DS-pipe shuffle.
__device__ __forceinline__ float xswap16(float x) {
#if __has_builtin(__builtin_amdgcn_permlanex16)
  unsigned u = __builtin_bit_cast(unsigned, x);
  unsigned r = __builtin_amdgcn_permlanex16(u, u, 0x76543210u, 0xfedcba98u, false, false);
  return __builtin_bit_cast(float, r);
#elif __has_builtin(__builtin_amdgcn_ds_swizzle)
  // group-of-32 swizzle, xor_mask=0x10, and_mask=0x1f  (SWAPX16)
  unsigned u = __builtin_bit_cast(unsigned, x);
  unsigned r = (unsigned)__builtin_amdgcn_ds_swizzle((int)u, 0x401F);
  return __builtin_bit_cast(float, r);
#else
  return __shfl_xor(x, 16, 32);
#endif
}

__device__ __forceinline__ void async_ld_b128(const float* g, float* l) {
#if USE_ASYNC_LOAD
  __builtin_amdgcn_global_load_async_to_lds_b128((AS1 v4i*)(void*)g, (AS3 v4i*)(void*)l, 0, 0);
#else
  asm volatile("global_load_async_to_lds_b128 %0, %1, off"
               :: "v"(lds_off(l)), "v"((unsigned long long)g) : "memory");
#endif
}

__device__ __forceinline__ void async_ld_b32(const float* g, float* l) {
#if USE_ASYNC_LOAD
  __builtin_amdgcn_global_load_async_to_lds_b32((AS1 int*)(void*)g, (AS3 int*)(void*)l, 0, 0);
#else
  asm volatile("global_load_async_to_lds_b32 %0, %1, off"
               :: "v"(lds_off(l)), "v"((unsigned long long)g) : "memory");
#endif
}

__device__ __forceinline__ void async_st_b32(float* g, const float* l) {
#if USE_ASYNC_STORE
  __builtin_amdgcn_global_store_async_from_lds_b32((AS1 int*)(void*)g, (AS3 int*)(void*)l, 0, 0);
#else
  asm volatile("global_store_async_from_lds_b32 %0, %1, off"
               :: "v"((unsigned long long)g), "v"(lds_off(l)) : "memory");
#endif
}

__device__ __forceinline__ void async_st_b128(float* g, const float* l) {
#if USE_ASYNC_STORE
  __builtin_amdgcn_global_store_async_from_lds_b128((AS1 v4i*)(void*)g, (AS3 v4i*)(void*)l, 0, 0);
#else
  asm volatile("global_store_async_from_lds_b128 %0, %1, off"
               :: "v"((unsigned long long)g), "v"(lds_off(l)) : "memory");
#endif
}

// One thread = one sphere. Block tile = 256 spheres.
// K is assumed a multiple of 256 (K = 2^20 here).
__global__ __launch_bounds__(TPB)
void fbank_cdna5_kernel(const float* __restrict__ sph,
                        const float* __restrict__ onesv,
                        const float* __restrict__ tet,
                        float* __restrict__ rot_out,
                        float* __restrict__ fb_out)
{
  __shared__ float smem[TPB * 20];  // 20 KB: input tile (1280f) -> rot stage (2304f) -> fb stage (5120f)
  const int tid = threadIdx.x;
  const long long blk = blockIdx.x;
  const float* gtile = sph + blk * (long long)(TPB * 5);

  // ---- async-stage input tile (5120 B) into LDS, fully coalesced ----
  async_ld_b128(gtile + tid * 4, smem + tid * 4);            // bytes [0,4096)
  async_ld_b32(gtile + 1024 + tid, smem + 1024 + tid);       // bytes [4096,5120)
  wait_async0();
  __syncthreads();

  const float s0 = smem[tid * 5 + 0];
  const float s1 = smem[tid * 5 + 1];
  const float s2 = smem[tid * 5 + 2];
  const float s3 = smem[tid * 5 + 3];
  const float s4 = smem[tid * 5 + 4];

  // ---- uniform constants: q = normalize(ones), Hq = I - (2/q.q) q q^T ----
  float q0 = onesv[0], q1 = onesv[1], q2 = onesv[2];
  const float rq = rsqrtf(q0 * q0 + q1 * q1 + q2 * q2);
  q0 *= rq; q1 *= rq; q2 *= rq;
  const float tq = 2.0f / (q0 * q0 + q1 * q1 + q2 * q2);
  const float h00 = 1.0f - tq * q0 * q0, h01 = -tq * q0 * q1, h02 = -tq * q0 * q2;
  const float h11 = 1.0f - tq * q1 * q1, h12 = -tq * q1 * q2;
  const float h22 = 1.0f - tq * q2 * q2;

  // ---- p = normalize(s/(s4+eps)); u = p + q; R0 = Hq - beta*(Hq u) u^T ----
  const float inw = 1.0f / (s4 + 1e-12f);
  const float c0 = s0 * inw, c1 = s1 * inw, c2 = s2 * inw;
  const float rc = rsqrtf(c0 * c0 + c1 * c1 + c2 * c2);
  const float u0 = c0 * rc + q0, u1 = c1 * rc + q1, u2 = c2 * rc + q2;
  const float bet = 2.0f / (u0 * u0 + u1 * u1 + u2 * u2);
  const float v0 = h00 * u0 + h01 * u1 + h02 * u2;
  const float v1 = h01 * u0 + h11 * u1 + h12 * u2;
  const float v2 = h02 * u0 + h12 * u1 + h22 * u2;
  const float bv0 = bet * v0, bv1 = bet * v1, bv2 = bet * v2;
  const float r00 = h00 - bv0 * u0, r01 = h01 - bv0 * u1, r02 = h02 - bv0 * u2;
  const float r10 = h01 - bv1 * u0, r11 = h11 - bv1 * u1, r12 = h12 - bv1 * u2;
  const float r20 = h02 - bv2 * u0, r21 = h12 - bv2 * u1, r22 = h22 - bv2 * u2;

  // w = R0 * s[0:3]
  const float w0 = r00 * s0 + r01 * s1 + r02 * s2;
  const float w1 = r10 * s0 + r11 * s1 + r12 * s2;
  const float w2 = r20 * s0 + r21 * s1 + r22 * s2;

#if USE_WMMA
  // ---- y_m = T_m * w via V_WMMA_F32_16X16X4_F32, two 16-sphere batches/wave ----
  // A (16x4): row 4m+r = row r of T_m (K=0..2 used, K=3 zero).
  //   lanes 0-15 hold K=0,1 in (A.x,A.y); lanes 16-31 hold K=2,3.
  const int lane = tid & 31;
  const bool hi = lane >= 16;
  const int mrow = lane & 15;
  const int mm = mrow >> 2, rr = mrow & 3;
  float a0 = 0.0f, a1 = 0.0f;
  if (rr < 3) {
    a0 = hi ? tet[mm * 9 + rr * 3 + 2] : tet[mm * 9 + rr * 3 + 0];
    a1 = hi ? 0.0f : tet[mm * 9 + rr * 3 + 1];
  }
  v2f A; A.x = a0; A.y = a1;

  // B (4x16): col n = w of sphere n. Lanes 0-15 carry (w0,w1); lanes 16-31 carry (w2,0).
  const float w0p = xswap16(w0);
  const float w1p = xswap16(w1);
  const float w2p = xswap16(w2);
  v2f B1, B2;
  B1.x = hi ? w2p : w0;  B1.y = hi ? 0.0f : w1;   // spheres of lanes 0-15
  B2.x = hi ? w2 : w0p;  B2.y = hi ? 0.0f : w1p;  // spheres of lanes 16-31
  v8f cz = {0.f, 0.f, 0.f, 0.f, 0.f, 0.f, 0.f, 0.f};
  v8f d1 = __builtin_amdgcn_wmma_f32_16x16x4_f32(false, A, false, B1, (short)0, cz, false, false);
  v8f d2 = __builtin_amdgcn_wmma_f32_16x16x4_f32(false, A, false, B2, (short)0, cz, false, false);

  // D rows are split across half-waves: swap with the partner lane (VALU permlane).
  const float p10 = xswap16(d1[0]), p11 = xswap16(d1[1]), p12 = xswap16(d1[2]);
  const float p14 = xswap16(d1[4]), p15 = xswap16(d1[5]), p16 = xswap16(d1[6]);
  const float p20 = xswap16(d2[0]), p21 = xswap16(d2[1]), p22 = xswap16(d2[2]);
  const float p24 = xswap16(d2[4]), p25 = xswap16(d2[5]), p26 = xswap16(d2[6]);

  const float ya0 = hi ? p20 : d1[0], ya1 = hi ? p21 : d1[1], ya2 = hi ? p22 : d1[2];  // m=0
  const float yb0 = hi ? p24 : d1[4], yb1 = hi ? p25 : d1[5], yb2 = hi ? p26 : d1[6];  // m=1
  const float yc0 = hi ? d2[0] : p10, yc1 = hi ? d2[1] : p11, yc2 = hi ? d2[2] : p12;  // m=2
  const float yd0 = hi ? d2[4] : p14, yd1 = hi ? d2[5] : p15, yd2 = hi ? d2[6] : p16;  // m=3
#else
  const float ya0 = tet[0] * w0 + tet[1] * w1 + tet[2] * w2;
  const float ya1 = tet[3] * w0 + tet[4] * w1 + tet[5] * w2;
  const float ya2 = tet[6] * w0 + tet[7] * w1 + tet[8] * w2;
  const float yb0 = tet[9] * w0 + tet[10] * w1 + tet[11] * w2;
  const float yb1 = tet[12] * w0 + tet[13] * w1 + tet[14] * w2;
  const float yb2 = tet[15] * w0 + tet[16] * w1 + tet[17] * w2;
  const float yc0 = tet[18] * w0 + tet[19] * w1 + tet[20] * w2;
  const float yc1 = tet[21] * w0 + tet[22] * w1 + tet[23] * w2;
  const float yc2 = tet[24] * w0 + tet[25] * w1 + tet[26] * w2;
  const float yd0 = tet[27] * w0 + tet[28] * w1 + tet[29] * w2;
  const float yd1 = tet[30] * w0 + tet[31] * w1 + tet[32] * w2;
  const float yd2 = tet[33] * w0 + tet[34] * w1 + tet[35] * w2;
#endif

  // z_m = R0^T y_m
  const float za0 = r00 * ya0 + r10 * ya1 + r20 * ya2;
  const float za1 = r01 * ya0 + r11 * ya1 + r21 * ya2;
  const float za2 = r02 * ya0 + r12 * ya1 + r22 * ya2;
  const float zb0 = r00 * yb0 + r10 * yb1 + r20 * yb2;
  const float zb1 = r01 * yb0 + r11 * yb1 + r21 * yb2;
  const float zb2 = r02 * yb0 + r12 * yb1 + r22 * yb2;
  const float zc0 = r00 * yc0 + r10 * yc1 + r20 * yc2;
  const float zc1 = r01 * yc0 + r11 * yc1 + r21 * yc2;
  const float zc2 = r02 * yc0 + r12 * yc1 + r22 * yc2;
  const float zd0 = r00 * yd0 + r10 * yd1 + r20 * yd2;
  const float zd1 = r01 * yd0 + r11 * yd1 + r21 * yd2;
  const float zd2 = r02 * yd0 + r12 * yd1 + r22 * yd2;

  // ---- stage rotations in LDS, flush coalesced via async stores ----
  __syncthreads();  // all input-tile reads done; smem is reusable
  {
    const int ro = tid * 9;
    smem[ro + 0] = r00; smem[ro + 1] = r01; smem[ro + 2] = r02;
    smem[ro + 3] = r10; smem[ro + 4] = r11; smem[ro + 5] = r12;
    smem[ro + 6] = r20; smem[ro + 7] = r21; smem[ro + 8] = r22;
  }
  __syncthreads();
  float* rbase = rot_out + blk * (long long)(TPB * 9);
#pragma unroll
  for (int j = 0; j < 9; ++j)
    async_st_b32(rbase + j * TPB + tid, smem + j * TPB + tid);
  wait_async0();
  __syncthreads();

  // ---- stage filter banks in LDS, flush coalesced (b128) ----
  {
    const int fo = tid * 20;
    smem[fo + 0] = za0; smem[fo + 1] = za1; smem[fo + 2] = za2; smem[fo + 3] = s3; smem[fo + 4] = s4;
    smem[fo + 5] = zb0; smem[fo + 6] = zb1; smem[fo + 7] = zb2; smem[fo + 8] = s3; smem[fo + 9] = s4;
    smem[fo + 10] = zc0; smem[fo + 11] = zc1; smem[fo + 12] = zc2; smem[fo + 13] = s3; smem[fo + 14] = s4;
    smem[fo + 15] = zd0; smem[fo + 16] = zd1; smem[fo + 17] = zd2; smem[fo + 18] = s3; smem[fo + 19] = s4;
  }
  __syncthreads();
  float* fbase = fb_out + blk * (long long)(TPB * 20);
#pragma unroll
  for (int j = 0; j < 5; ++j)
    async_st_b128(fbase + j * 1024 + tid * 4, smem + j * 1024 + tid * 4);
  // s_endpgm performs an implicit wait-idle, draining the async stores.
}

extern "C" void kernel_launch(void* const* d_in, const int* in_sizes, int n_in,
                              void* d_out, int out_size, void* d_ws, size_t ws_size,
                              hipStream_t stream) {
  (void)n_in; (void)out_size; (void)d_ws; (void)ws_size;
  const float* sph = (const float*)d_in[0];
  const float* onesv = (const float*)d_in[1];
  const float* tet = (const float*)d_in[2];
  const int K = in_sizes[0] / 5;  // (K, D+2) with D+2 == 5
  float* rot_out = (float*)d_out;                     // K*9 floats
  float* fb_out = (float*)d_out + (size_t)K * 9;      // K*4*5 floats
  const int nblocks = K / TPB;                        // K = 2^20 -> 4096
  fbank_cdna5_kernel<<<dim3(nblocks), dim3(TPB), 0, stream>>>(sph, onesv, tet, rot_out, fb_out);
}